// BatchAllTripletLoss_14525579395273
// MI455X (gfx1250) — compile-verified
//
#include <hip/hip_runtime.h>
#include <hip/hip_bf16.h>
#include <math.h>

#define B_SZ 384
#define D_SZ 256
#define NTILE 24        // 384 / 16
#define MARGIN_F 5.0f

typedef __attribute__((ext_vector_type(2))) float v2f;
typedef __attribute__((ext_vector_type(8))) float v8f;

// ---------------------------------------------------------------------------
// Kernel 0: row squared-norms (one wave32 per row) + zero global accumulators
// ---------------------------------------------------------------------------
__global__ __launch_bounds__(32) void tl_norms_kernel(const float* __restrict__ emb,
                                                      float* __restrict__ norms,
                                                      float* __restrict__ accum) {
    const int row  = blockIdx.x;
    const int lane = threadIdx.x;
    const float* r = emb + (size_t)row * D_SZ;
    float s = 0.0f;
    #pragma unroll
    for (int k = lane; k < D_SZ; k += 32) {
        float v = r[k];
        s = fmaf(v, v, s);
    }
    #pragma unroll
    for (int off = 16; off > 0; off >>= 1)
        s += __shfl_xor(s, off, 32);
    if (lane == 0) norms[row] = s;
    if (row == 0 && lane == 0) { accum[0] = 0.0f; accum[1] = 0.0f; }
}

// ---------------------------------------------------------------------------
// Kernel 1: pairwise Euclidean distance tiles via V_WMMA_F32_16X16X4_F32.
// One wave32 per 16x16 output tile; K=256 consumed in 64 steps of 4.
//
// A-matrix (16x4 f32) lane layout (CDNA5 ISA 7.12.2):
//   lanes 0-15 : M=lane,    VGPR0=K0, VGPR1=K1
//   lanes 16-31: M=lane-16, VGPR0=K2, VGPR1=K3
// => per lane: two consecutive floats at column k0 + 2*(lane>>4).  B mirrors
// this with N in place of M (B = emb^T, so B[k][n] = emb[n][k]).
// C/D (16x16 f32): VGPR r holds M = r + 8*(lane>>4), N = lane&15.
// ---------------------------------------------------------------------------
__global__ __launch_bounds__(32) void tl_pdist_wmma_kernel(const float* __restrict__ emb,
                                                           const float* __restrict__ norms,
                                                           float* __restrict__ pdist) {
    const int tm   = blockIdx.x;       // tile row  (M)
    const int tn   = blockIdx.y;       // tile col  (N)
    const int lane = threadIdx.x;
    const int r16  = lane & 15;
    const int hi   = lane >> 4;        // 0 or 1

    const float* arow = emb + (size_t)(tm * 16 + r16) * D_SZ;
    const float* brow = emb + (size_t)(tn * 16 + r16) * D_SZ;

    v8f c = {};
    #pragma unroll 4
    for (int k0 = 0; k0 < D_SZ; k0 += 4) {
        const float2 av = *(const float2*)(arow + k0 + 2 * hi);
        const float2 bv = *(const float2*)(brow + k0 + 2 * hi);
        v2f a; a.x = av.x; a.y = av.y;
        v2f b; b.x = bv.x; b.y = bv.y;
        // 8 args: (neg_a, A, neg_b, B, c_mod, C, reuse_a, reuse_b)
        c = __builtin_amdgcn_wmma_f32_16x16x4_f32(
                /*neg_a=*/false, a, /*neg_b=*/false, b,
                /*c_mod=*/(short)0, c, /*reuse_a=*/false, /*reuse_b=*/false);
    }

    // dot -> distance: d2 = |x|^2 + |y|^2 - 2<x,y>, clamp, sqrt (0 stays 0)
    const int gj = tn * 16 + r16;
    const float nj = norms[gj];
    #pragma unroll
    for (int r = 0; r < 8; ++r) {
        const int gi = tm * 16 + r + 8 * hi;
        float d2 = norms[gi] + nj - 2.0f * c[r];
        d2 = fmaxf(d2, 0.0f);
        pdist[(size_t)gi * B_SZ + gj] = (d2 > 0.0f) ? sqrtf(d2) : 0.0f;
    }
}

// ---------------------------------------------------------------------------
// Kernel 2: batch-all triplet reduction, divergence-free formulation.
// One block per anchor i.  Positives for anchor i (label match, j!=i) are
// compacted into an LDS list (~B/NCLASSES entries).  Then EVERY thread owns
// one candidate negative k and loops only over the short positive list, so
// all 32 lanes of each wave stay active (vs. ~6/384 in the naive j-owns-work
// layout).  Mask algebra: label[j]==label[i], i!=j, label[k]!=label[i]
// => i,j,k automatically distinct.
// ---------------------------------------------------------------------------
__global__ __launch_bounds__(B_SZ) void tl_triplet_kernel(const float* __restrict__ pdist,
                                                          const int*   __restrict__ labels,
                                                          float* __restrict__ accum) {
    __shared__ float srow[B_SZ];     // pdist row of anchor i
    __shared__ int   slab[B_SZ];     // labels
    __shared__ float spos[B_SZ];     // d(i, j) for compacted positives j
    __shared__ int   snpos;
    __shared__ float sred[2];

    const int i = blockIdx.x;
    const int t = threadIdx.x;

    srow[t] = pdist[(size_t)i * B_SZ + t];
    slab[t] = labels[t];
    if (t == 0) snpos = 0;
    if (t < 2) sred[t] = 0.0f;
    __syncthreads();

    const int li = slab[i];

    // Compact positive anchors' distances into spos[0..npos)
    if (t != i && slab[t] == li) {
        int idx = atomicAdd(&snpos, 1);   // LDS atomic (ds_add_rtn_u32)
        spos[idx] = srow[t];
    }
    __syncthreads();

    const int npos = snpos;
    float sum = 0.0f, cnt = 0.0f;
    if (slab[t] != li) {                  // thread t is a valid negative k
        const float dik = srow[t];
        const float base = MARGIN_F - dik;
        for (int p = 0; p < npos; ++p) {
            const float tl = spos[p] + base;   // d_ij - d_ik + margin
            sum += fmaxf(tl, 0.0f);
            cnt += (tl > 1e-16f) ? 1.0f : 0.0f;
        }
    }
    atomicAdd(&sred[0], sum);   // LDS float atomic (ds_add_f32)
    atomicAdd(&sred[1], cnt);
    __syncthreads();
    if (t == 0) {
        atomicAdd(&accum[0], sred[0]);
        atomicAdd(&accum[1], sred[1]);
    }
}

// ---------------------------------------------------------------------------
// Kernel 3: finalize scalar loss
// ---------------------------------------------------------------------------
__global__ void tl_finalize_kernel(const float* __restrict__ accum,
                                   float* __restrict__ out) {
    out[0] = accum[0] / (accum[1] + 1e-16f);
}

// ---------------------------------------------------------------------------
extern "C" void kernel_launch(void* const* d_in, const int* in_sizes, int n_in,
                              void* d_out, int out_size, void* d_ws, size_t ws_size,
                              hipStream_t stream) {
    const float* emb    = (const float*)d_in[0];   // [384,256] f32
    const int*   labels = (const int*)d_in[1];     // [384] int
    float*       out    = (float*)d_out;           // scalar loss

    float* ws     = (float*)d_ws;
    float* norms  = ws;                              // 384 floats
    float* pdist  = ws + B_SZ;                       // 384*384 floats
    float* accum  = ws + B_SZ + B_SZ * B_SZ;         // 2 floats (sum, count)

    tl_norms_kernel<<<B_SZ, 32, 0, stream>>>(emb, norms, accum);
    tl_pdist_wmma_kernel<<<dim3(NTILE, NTILE), 32, 0, stream>>>(emb, norms, pdist);
    tl_triplet_kernel<<<B_SZ, B_SZ, 0, stream>>>(pdist, labels, accum);
    tl_finalize_kernel<<<1, 1, 0, stream>>>(accum, out);
}